// APPNP_33449205301778
// MI455X (gfx1250) — compile-verified
//
#include <hip/hip_runtime.h>

#define N_NODES 100000
#define N_EDGES 1600000
#define IN_F 256
#define HID 128
#define OUT_F 48
#define ALPHA 0.01f
#define KSTEPS 10
#define DROP_SCALE (1.0f / (1.0f + 1e-5f))
#define LDSP (HID + 4)   // padded LDS row stride (floats) to avoid bank conflicts

typedef __attribute__((ext_vector_type(2))) float v2f;
typedef __attribute__((ext_vector_type(8))) float v8f;

// ---------------------------------------------------------------------------
// Fused MLP: out = relu(x @ W1 + b1) @ W2 + b2
// One wave computes one 16-node tile. fp32 WMMA (V_WMMA_F32_16X16X4_F32).
// ---------------------------------------------------------------------------
__global__ __launch_bounds__(128) void mlp_wmma_kernel(
    const float* __restrict__ x,
    const float* __restrict__ W1, const float* __restrict__ b1,
    const float* __restrict__ W2, const float* __restrict__ b2,
    float* __restrict__ out)
{
    __shared__ float lds_h1[4][16 * LDSP];   // per-wave 16x128 tile (padded)

    const int wave  = threadIdx.x >> 5;
    const int lane  = threadIdx.x & 31;
    const int tile  = blockIdx.x * 4 + wave;          // 16-node tile index
    if (tile * 16 >= N_NODES) return;                 // wave-uniform guard (EXEC stays all-1s)

    const int m     = lane & 15;                      // row (A) / col (B,C,D) index
    const int khalf = lane >> 4;                      // 0 or 1
    const int node0 = tile * 16;

    // ----- Layer 1: 16x256 @ 256x128 -> 16x128, 8 N-tiles, K in chunks of 4
    v8f acc1[8] = {};
    const float* xrow = x + (size_t)(node0 + m) * IN_F;
    for (int k0 = 0; k0 < IN_F; k0 += 4) {
        const int kb = k0 + khalf * 2;
        v2f a;
        a.x = xrow[kb];
        a.y = xrow[kb + 1];
#pragma unroll
        for (int t = 0; t < 8; ++t) {
            const int n = t * 16 + m;
            v2f b;
            b.x = W1[(size_t)kb * HID + n];
            b.y = W1[(size_t)(kb + 1) * HID + n];
            acc1[t] = __builtin_amdgcn_wmma_f32_16x16x4_f32(
                false, a, false, b, (short)0, acc1[t], false, false);
        }
    }

    // bias + relu, write tile to LDS (row-major, padded)
    float* hl = lds_h1[wave];
#pragma unroll
    for (int t = 0; t < 8; ++t) {
        const int col = t * 16 + m;
        const float bias = b1[col];
#pragma unroll
        for (int j = 0; j < 8; ++j) {
            const int row = j + 8 * khalf;
            const float v = acc1[t][j] + bias;
            hl[row * LDSP + col] = v > 0.f ? v : 0.f;
        }
    }
    // same-wave LDS store->load ordering is handled by DScnt waits (in-order per wave)

    // ----- Layer 2: 16x128 @ 128x48 -> 16x48, 3 N-tiles
    v8f acc2[3] = {};
    for (int k0 = 0; k0 < HID; k0 += 4) {
        const int kb = k0 + khalf * 2;
        v2f a;
        a.x = hl[m * LDSP + kb];
        a.y = hl[m * LDSP + kb + 1];
#pragma unroll
        for (int t = 0; t < 3; ++t) {
            const int n = t * 16 + m;
            v2f b;
            b.x = W2[(size_t)kb * OUT_F + n];
            b.y = W2[(size_t)(kb + 1) * OUT_F + n];
            acc2[t] = __builtin_amdgcn_wmma_f32_16x16x4_f32(
                false, a, false, b, (short)0, acc2[t], false, false);
        }
    }

    // bias + store
#pragma unroll
    for (int t = 0; t < 3; ++t) {
        const int col = t * 16 + m;
        const float bias = b2[col];
#pragma unroll
        for (int j = 0; j < 8; ++j) {
            const int row = j + 8 * khalf;
            out[(size_t)(node0 + row) * OUT_F + col] = acc2[t][j] + bias;
        }
    }
}

// ---------------------------------------------------------------------------
// dst = ALPHA * src  (blend-init for one propagation step)
// ---------------------------------------------------------------------------
__global__ __launch_bounds__(256) void scale_init_kernel(
    float* __restrict__ dst, const float* __restrict__ src, int n)
{
    const int i = blockIdx.x * 256 + threadIdx.x;
    if (i < n) dst[i] = ALPHA * src[i];
}

// ---------------------------------------------------------------------------
// Scatter: acc[row] += (1-ALPHA) * DROP_SCALE * val * h[col]
// 12 threads per edge, float4 (16B) gather, 4 native fp32 global atomics
// (no-return -> STOREcnt path, resolved in L2 atomic units; h buffers and
// edge lists are all L2-resident at 192 MB).
// Inline asm guarantees GLOBAL_ATOMIC_ADD_F32 (no CAS-loop fallback).
// ---------------------------------------------------------------------------
__global__ __launch_bounds__(256) void edge_scatter_kernel(
    const int* __restrict__ rows, const int* __restrict__ cols,
    const float* __restrict__ vals,
    const float* __restrict__ hsrc, float* __restrict__ acc)
{
    const long long tid = (long long)blockIdx.x * 256 + threadIdx.x;
    if (tid >= (long long)N_EDGES * 12) return;
    const int e = (int)(tid / 12);
    const int c = (int)(tid % 12);
    const int f = c * 4;

    const int r   = rows[e];
    const int col = cols[e];
    const float w = vals[e] * (DROP_SCALE * (1.0f - ALPHA));

    const float4 hv = *(const float4*)(hsrc + (size_t)col * OUT_F + f);
    float* dst = acc + (size_t)r * OUT_F + f;

    const float a0 = w * hv.x;
    const float a1 = w * hv.y;
    const float a2 = w * hv.z;
    const float a3 = w * hv.w;

    asm volatile(
        "global_atomic_add_f32 %0, %1, off\n\t"
        "global_atomic_add_f32 %0, %2, off offset:4\n\t"
        "global_atomic_add_f32 %0, %3, off offset:8\n\t"
        "global_atomic_add_f32 %0, %4, off offset:12"
        :
        : "v"(dst), "v"(a0), "v"(a1), "v"(a2), "v"(a3)
        : "memory");
}

// ---------------------------------------------------------------------------
extern "C" void kernel_launch(void* const* d_in, const int* in_sizes, int n_in,
                              void* d_out, int out_size, void* d_ws, size_t ws_size,
                              hipStream_t stream)
{
    (void)in_sizes; (void)n_in; (void)out_size; (void)ws_size;

    const float* x    = (const float*)d_in[0];
    const int*   rows = (const int*)  d_in[1];
    const int*   cols = (const int*)  d_in[2];
    const float* vals = (const float*)d_in[3];
    const float* W1   = (const float*)d_in[4];
    const float* b1   = (const float*)d_in[5];
    const float* W2   = (const float*)d_in[6];
    const float* b2   = (const float*)d_in[7];

    float* out  = (float*)d_out;
    float* buf0 = (float*)d_ws;          // one N_NODES x OUT_F scratch buffer

    // MLP writes directly into d_out; with 10 steps and 2 buffers, parity
    // (even steps -> buf0, odd steps -> d_out) makes step 9 land in d_out.
    const int n_tiles = N_NODES / 16;                    // 6250, exact
    mlp_wmma_kernel<<<(n_tiles + 3) / 4, 128, 0, stream>>>(x, W1, b1, W2, b2, out);

    const int nElem = N_NODES * OUT_F;
    const long long edgeThreads = (long long)N_EDGES * 12;
    const int edgeBlocks = (int)((edgeThreads + 255) / 256);

    const float* cur = out;
    for (int step = 0; step < KSTEPS; ++step) {
        float* dst = (step & 1) ? out : buf0;
        scale_init_kernel<<<(nElem + 255) / 256, 256, 0, stream>>>(dst, cur, nElem);
        edge_scatter_kernel<<<edgeBlocks, 256, 0, stream>>>(rows, cols, vals, cur, dst);
        cur = dst;
    }
}